// interactionModule_85229331022334
// MI455X (gfx1250) — compile-verified
//
#include <hip/hip_runtime.h>

// Reference constants (N=100000, E=3200000, D=2, C=1, R_C=1, P=2, GAMMA=0.1, EPS=1e-12)
#define TILE 256
#define GAMMA 0.1f
#define EPS_F 1e-12f

// ---------------------------------------------------------------------------
// Kernel 1: out = -gamma * v   (float2-vectorized elementwise init)
// ---------------------------------------------------------------------------
__global__ __launch_bounds__(256) void init_damp_kernel(const float2* __restrict__ v,
                                                        float2* __restrict__ out,
                                                        int n) {
    int i = blockIdx.x * blockDim.x + threadIdx.x;
    if (i < n) {
        float2 vv = v[i];
        out[i] = make_float2(-GAMMA * vv.x, -GAMMA * vv.y);
    }
}

// ---------------------------------------------------------------------------
// Kernel 2: per-edge spring message + scatter-add via global f32 atomics.
//   m = -2*(|dr|-1) * dr / max(|dr|, eps);  out[dst] += m
// Edge indices for each 256-edge tile are staged into LDS with the CDNA5
// async copy path (global_load_async_to_lds_b128 + s_wait_asynccnt).
// ---------------------------------------------------------------------------
__global__ __launch_bounds__(TILE) void edge_scatter_kernel(const float2* __restrict__ x,
                                                            const int* __restrict__ src,
                                                            const int* __restrict__ dst,
                                                            float* __restrict__ out,
                                                            int E) {
    __shared__ int s_src[TILE];
    __shared__ int s_dst[TILE];

    const int tid   = threadIdx.x;
    const int tile0 = blockIdx.x * TILE;

    int si, di;
    bool active = true;

    if (tile0 + TILE <= E) {
        // --- async stage: 64 lanes x 16B cover 256 ints per index array ---
        if (tid < 128) {
            const int lane = tid & 63;
            const int* g = (tid < 64) ? (src + tile0 + lane * 4)
                                      : (dst + tile0 + lane * 4);
            int* l = (tid < 64) ? (s_src + lane * 4) : (s_dst + lane * 4);
            // LDS flat addresses carry the byte offset in addr[31:0] (ISA 10.2):
            unsigned lds_off = (unsigned)(unsigned long long)(void*)l;
            unsigned long long ga = (unsigned long long)(const void*)g;
            asm volatile("global_load_async_to_lds_b128 %0, %1, off"
                         :: "v"(lds_off), "v"(ga)
                         : "memory");
        }
        asm volatile("s_wait_asynccnt 0" ::: "memory");
        __syncthreads();
        si = s_src[tid];
        di = s_dst[tid];
    } else {
        // tail tile (not taken for E = 3.2M, kept for generality)
        int e = tile0 + tid;
        active = (e < E);
        si = active ? src[e] : 0;
        di = active ? dst[e] : 0;
    }

    if (active) {
        float2 xs = x[si];          // gathers hit L2/WGP$ (table is 800 KB)
        float2 xd = x[di];
        float dx = xd.x - xs.x;
        float dy = xd.y - xs.y;
        float r  = sqrtf(dx * dx + dy * dy);
        float inv = 1.0f / fmaxf(r, EPS_F);           // F.normalize eps
        float f   = -2.0f * (r - 1.0f) * inv;         // -C*P*(r-R_C)^(P-1) / r
        atomicAdd(&out[2 * di],     f * dx);
        atomicAdd(&out[2 * di + 1], f * dy);
    }
}

// ---------------------------------------------------------------------------
// Harness entry point.
// Inputs (setup_inputs order): x[N*2] f32, v[N*2] f32, src[E] i32, dst[E] i32.
// Output: [N*2] f32.
// ---------------------------------------------------------------------------
extern "C" void kernel_launch(void* const* d_in, const int* in_sizes, int n_in,
                              void* d_out, int out_size, void* d_ws, size_t ws_size,
                              hipStream_t stream) {
    const float* x = (const float*)d_in[0];
    const float* v = (const float*)d_in[1];
    const int* src = (const int*)d_in[2];
    const int* dst = (const int*)d_in[3];
    float* out = (float*)d_out;

    const int N = in_sizes[0] / 2;   // 100000
    const int E = in_sizes[2];       // 3200000

    init_damp_kernel<<<(N + 255) / 256, 256, 0, stream>>>(
        (const float2*)v, (float2*)out, N);

    edge_scatter_kernel<<<(E + TILE - 1) / TILE, TILE, 0, stream>>>(
        (const float2*)x, src, dst, out, E);
}